// CustomLoss_57123065037580
// MI455X (gfx1250) — compile-verified
//
#include <hip/hip_runtime.h>
#include <hip/hip_bf16.h>

#define NUM_CLASSES 21
#define ROWF        25            // 21 logits + 4 box regs
#define TILE_ANCH   128           // anchors per wave-tile
#define WAVES_PER_WG 8
#define WG_THREADS  256
#define TILE_DW     (TILE_ANCH * ROWF)   // 3200 dwords = 12800 B
#define TILE_BYTES  (TILE_DW * 4)
#define CHUNKS      (TILE_BYTES / 16 / 32) // 25 b128 chunks per lane-sweep

// ---------------- CDNA5 async-tensor staging (CDNA5 ISA) -------
// Per lane: copies 16 bytes from global address g to LDS address lds; tracked on ASYNCcnt.
static __device__ __forceinline__ void async_b128(unsigned lds, const void* g) {
  asm volatile("global_load_async_to_lds_b128 %0, %1, off" :: "v"(lds), "v"(g));
}
static __device__ __forceinline__ void wait_async0() {
  asm volatile("s_wait_asynccnt 0" ::: "memory");
}
// Deterministic fixed-point accumulator: value * 2^32 as u64 (fl in [0, ~32))
static __device__ __forceinline__ unsigned long long f2fx(float x) {
  return (unsigned long long)((double)x * 4294967296.0);
}

// ws scalar slots
// scal[0]=posCount scal[1]=negCount scal[3]=negWriteIdx scal[4]=radixPrefix
// scal[5]=radixRemaining scal[6]=Tbits scal[7]=tieTake scal[8]=k
// sums[0]=posFlFx sums[1]=regFx sums[2]=negAboveFx

__global__ void k_init(unsigned* scal, unsigned long long* sums, unsigned* hist) {
  const int t = threadIdx.x;
  if (t < 16)  scal[t] = 0u;
  if (t < 8)   sums[t] = 0ull;
  if (t < 256) hist[t] = 0u;
}

__global__ void __launch_bounds__(WG_THREADS)
k1_main(const float* __restrict__ yp, const float* __restrict__ yb,
        const float* __restrict__ anc,
        unsigned* __restrict__ scal, unsigned long long* __restrict__ sums,
        unsigned* __restrict__ negfl, unsigned negCap,
        int Ntot, int nAnchors, int numTiles)
{
  extern __shared__ float smem[];
  const int w = threadIdx.x >> 5;
  const int l = threadIdx.x & 31;
  float* sP = smem + w * TILE_DW;
  float* sB = smem + WAVES_PER_WG * TILE_DW + w * TILE_DW;
  const unsigned ldsP = (unsigned)(size_t)sP;   // low 32 bits = LDS byte address
  const unsigned ldsB = (unsigned)(size_t)sB;
  const int wavesTotal = gridDim.x * WAVES_PER_WG;
  const long long maxByte = (long long)Ntot * (ROWF * 4);
  const bool ancPow2 = (nAnchors & (nAnchors - 1)) == 0;

  unsigned posC = 0, negC = 0;
  unsigned long long posFx = 0, regFx = 0;

  for (int tile = blockIdx.x * WAVES_PER_WG + w; tile < numTiles; tile += wavesTotal) {
    const long long tb = (long long)tile * TILE_BYTES;
    const char* gp = (const char*)yp + tb;
    const char* gb = (const char*)yb + tb;
    #pragma unroll
    for (int i = 0; i < CHUNKS; ++i) {
      const int off = (i * 32 + l) * 16;
      const bool ok = (tb + off + 16) <= maxByte;       // tail-safe clamp
      async_b128(ldsP + off, ok ? (gp + off) : (const char*)yp);
    }
    #pragma unroll
    for (int i = 0; i < CHUNKS; ++i) {
      const int off = (i * 32 + l) * 16;
      const bool ok = (tb + off + 16) <= maxByte;
      async_b128(ldsB + off, ok ? (gb + off) : (const char*)yb);
    }
    wait_async0();

    #pragma unroll
    for (int j = 0; j < TILE_ANCH / 32; ++j) {
      const int a = l + 32 * j;
      const long long g = (long long)tile * TILE_ANCH + a;
      const bool valid = g < (long long)Ntot;
      const float* cp = sP + a * ROWF;   // predictions (logits + regs)
      const float* ch = sB + a * ROWF;   // targets (one-hot + regs)

      // argmax over one-hot target + max over logits (fused sweep)
      float m = ch[0]; int gt = 0;
      float mx = cp[0];
      #pragma unroll
      for (int c = 1; c < NUM_CLASSES; ++c) {
        const float v = ch[c]; if (v > m) { m = v; gt = c; }
        mx = fmaxf(mx, cp[c]);
      }
      float s = 0.f;
      #pragma unroll
      for (int c = 0; c < NUM_CLASSES; ++c) s += __expf(cp[c] - mx);
      const float logpt = cp[gt] - mx - __logf(s);
      const float pt    = __expf(logpt);
      const float alpha = (gt == 0) ? 0.25f : 1.0f;
      const float om    = 1.f - pt;
      const float fl    = -(om * om) * (logpt * alpha);   // >= 0

      const bool pos = valid && (gt != 0) && (m == 1.0f);
      const bool neg = valid && (gt == 0);
      posC += pos ? 1u : 0u;
      negC += neg ? 1u : 0u;

      if (pos) {
        posFx += f2fx(fl);
        const unsigned gu = (unsigned)g;
        const unsigned ai = ancPow2 ? (gu & (unsigned)(nAnchors - 1))
                                    : (gu % (unsigned)nAnchors);
        const float4 A = ((const float4*)anc)[ai];
        const float wa = A.z - A.x, ha = A.w - A.y;
        const float cxa = A.x + 0.5f * wa, cya = A.y + 0.5f * ha;
        // decode prediction
        const float cx = cp[21] * wa + cxa, cy = cp[22] * ha + cya;
        const float W  = __expf(cp[23]) * wa, H = __expf(cp[24]) * ha;
        const float px1 = cx - 0.5f * W, py1 = cy - 0.5f * H;
        const float px2 = cx + 0.5f * W, py2 = cy + 0.5f * H;
        // decode target
        const float tcx = ch[21] * wa + cxa, tcy = ch[22] * ha + cya;
        const float tW  = __expf(ch[23]) * wa, tH = __expf(ch[24]) * ha;
        const float tx1 = tcx - 0.5f * tW, ty1 = tcy - 0.5f * tH;
        const float tx2 = tcx + 0.5f * tW, ty2 = tcy + 0.5f * tH;
        float ix = fminf(px2, tx2) - fmaxf(px1, tx1); ix = fmaxf(ix, 0.f);
        float iy = fminf(py2, ty2) - fmaxf(py1, ty1); iy = fmaxf(iy, 0.f);
        const float inter = ix * iy;
        const float a1 = fmaxf(px2 - px1, 0.f) * fmaxf(py2 - py1, 0.f);
        const float a2 = fmaxf(tx2 - tx1, 0.f) * fmaxf(ty2 - ty1, 0.f);
        const float iou = inter / (a1 + a2 - inter + 1e-8f);
        regFx += f2fx(1.f - iou);
      }

      // wave32 ballot compaction of negative focal values
      const unsigned mask = (unsigned)__ballot(neg ? 1 : 0);
      const unsigned cnt  = __popc(mask);
      if (cnt) {
        unsigned base = 0;
        if (l == 0) base = atomicAdd(&scal[3], cnt);
        base = __shfl(base, 0);
        if (neg) {
          const unsigned idx = base + (unsigned)__popc(mask & ((1u << l) - 1u));
          if (idx < negCap) negfl[idx] = __float_as_uint(fl);
        }
      }
    }
  }

  // wave32 tree reduction, then one atomic per wave per quantity (order-independent)
  for (int off = 16; off; off >>= 1) {
    posC  += __shfl_down(posC,  off);
    negC  += __shfl_down(negC,  off);
    posFx += __shfl_down(posFx, off);
    regFx += __shfl_down(regFx, off);
  }
  if (l == 0) {
    if (posC)  atomicAdd(&scal[0], posC);
    if (negC)  atomicAdd(&scal[1], negC);
    if (posFx) atomicAdd(&sums[0], posFx);
    if (regFx) atomicAdd(&sums[1], regFx);
  }
}

__global__ void k_prep(unsigned* scal) {
  if (threadIdx.x == 0) {
    const unsigned np = scal[0], nn = scal[1];
    const unsigned long long k3 = 3ull * (unsigned long long)np;
    const unsigned k = ((unsigned long long)nn < k3) ? nn : (unsigned)k3;
    scal[8] = k;            // k
    scal[4] = 0u;           // radix prefix
    scal[5] = k;            // remaining
    scal[6] = 0xFFFFFFFFu;  // T default (selects nothing)
    scal[7] = 0u;           // tie take default
  }
}

__global__ void k_hist(const unsigned* __restrict__ negfl,
                       const unsigned* __restrict__ scal,
                       unsigned* __restrict__ hist,
                       int shift, unsigned highMask, unsigned negCap)
{
  __shared__ unsigned h[256];
  if (threadIdx.x < 256) h[threadIdx.x] = 0u;
  __syncthreads();
  const unsigned rem = scal[5];
  if (rem) {
    unsigned n = scal[1]; if (n > negCap) n = negCap;
    const unsigned prefix = scal[4];
    const unsigned stride = gridDim.x * blockDim.x;
    for (unsigned i = blockIdx.x * blockDim.x + threadIdx.x; i < n; i += stride) {
      const unsigned v = negfl[i];
      if (((v ^ prefix) & highMask) == 0u)
        atomicAdd(&h[(v >> shift) & 255u], 1u);
    }
  }
  __syncthreads();
  if (threadIdx.x < 256 && h[threadIdx.x]) atomicAdd(&hist[threadIdx.x], h[threadIdx.x]);
}

__global__ void k_scan(unsigned* scal, unsigned* hist, int shift) {
  __shared__ unsigned h[256];
  h[threadIdx.x] = hist[threadIdx.x];
  hist[threadIdx.x] = 0u;               // ready for next pass
  __syncthreads();
  if (threadIdx.x == 0) {
    const unsigned rem = scal[5];
    if (rem) {
      unsigned c = 0; int b = 255;
      for (; b > 0; --b) { const unsigned hb = h[b]; if (c + hb >= rem) break; c += hb; }
      scal[4] |= ((unsigned)b) << shift;
      scal[5]  = rem - c;
      if (shift == 0) { scal[6] = scal[4]; scal[7] = rem - c; }
    }
  }
}

__global__ void k_sumabove(const unsigned* __restrict__ negfl,
                           const unsigned* __restrict__ scal,
                           unsigned long long* __restrict__ sums, unsigned negCap)
{
  unsigned n = scal[1]; if (n > negCap) n = negCap;
  const unsigned T = scal[6];
  unsigned long long acc = 0;
  const unsigned stride = gridDim.x * blockDim.x;
  for (unsigned i = blockIdx.x * blockDim.x + threadIdx.x; i < n; i += stride) {
    const unsigned v = negfl[i];
    if (v > T) acc += f2fx(__uint_as_float(v));
  }
  for (int off = 16; off; off >>= 1) acc += __shfl_down(acc, off);
  if ((threadIdx.x & 31) == 0 && acc) atomicAdd(&sums[2], acc);
}

__global__ void k_final(const unsigned* scal, const unsigned long long* sums, float* out) {
  if (threadIdx.x != 0) return;
  const double inv = 1.0 / 4294967296.0;
  const unsigned np = scal[0], nn = scal[1], k = scal[8], T = scal[6], r = scal[7];
  const double posS  = (double)sums[0] * inv;
  const double regS  = (double)sums[1] * inv;
  const double negAb = (double)sums[2] * inv;
  const double tieV  = r ? (double)__uint_as_float(T) : 0.0;
  const double negSum = negAb + (double)r * tieV;
  const double pos_loss = posS / (double)(np > 0 ? np : 1u);
  const double neg_loss = k ? (negSum / (double)k) : 0.0;
  const double cls = (np > 0 && nn > 0) ? (pos_loss + neg_loss) : 0.0;
  const double reg = np ? (regS / (double)np) : 0.0;
  out[0] = (float)(0.5 * cls + 0.5 * reg);
  out[1] = (float)cls;
  out[2] = (float)reg;
}

extern "C" void kernel_launch(void* const* d_in, const int* in_sizes, int n_in,
                              void* d_out, int out_size, void* d_ws, size_t ws_size,
                              hipStream_t stream)
{
  const float* yp  = (const float*)d_in[0];
  const float* yb  = (const float*)d_in[1];
  const float* anc = (const float*)d_in[2];
  const int Ntot     = in_sizes[0] / (NUM_CLASSES + 4);   // 2,097,152
  const int nAnchors = in_sizes[2] / 4;                   // 32,768
  const int numTiles = (Ntot + TILE_ANCH - 1) / TILE_ANCH;

  unsigned*           scal  = (unsigned*)d_ws;
  unsigned long long* sums  = (unsigned long long*)((char*)d_ws + 64);
  unsigned*           hist  = (unsigned*)((char*)d_ws + 128);
  unsigned*           negfl = (unsigned*)((char*)d_ws + 1280);
  const unsigned negCap = (ws_size > 1280) ? (unsigned)((ws_size - 1280) / 4) : 0u;

  k_init<<<1, 256, 0, stream>>>(scal, sums, hist);

  int wgs = (numTiles + WAVES_PER_WG - 1) / WAVES_PER_WG;
  if (wgs > 4096) wgs = 4096;
  if (wgs < 1) wgs = 1;
  const size_t smemBytes = (size_t)2 * WAVES_PER_WG * TILE_DW * sizeof(float); // 204,800 B (<320 KB)
  k1_main<<<wgs, WG_THREADS, smemBytes, stream>>>(yp, yb, anc, scal, sums, negfl, negCap,
                                                  Ntot, nAnchors, numTiles);

  k_prep<<<1, 32, 0, stream>>>(scal);
  for (int shift = 24; shift >= 0; shift -= 8) {
    const unsigned highMask = (shift == 24) ? 0u : (0xFFFFFFFFu << (shift + 8));
    k_hist<<<256, 256, 0, stream>>>(negfl, scal, hist, shift, highMask, negCap);
    k_scan<<<1, 256, 0, stream>>>(scal, hist, shift);
  }
  k_sumabove<<<256, 256, 0, stream>>>(negfl, scal, sums, negCap);
  k_final<<<1, 32, 0, stream>>>(scal, sums, (float*)d_out);
}